// ULayer_75471165325354
// MI455X (gfx1250) — compile-verified
//
#include <hip/hip_runtime.h>
#include <math.h>

#define MODES 128
#define NTOT (MODES * MODES)

typedef float v2f __attribute__((ext_vector_type(2)));
typedef float v8f __attribute__((ext_vector_type(8)));

// ---------------------------------------------------------------------------
// Kernel 1: build U = (scan of Givens rotations) applied to diag(exp(i*diag)).
// Rows of U are independent under the column-pair update, so one thread per
// row runs the whole sequential scan with no synchronization. U lives in LDS
// (2 * 64KB planes), final result written to global scratch row-major.
// ---------------------------------------------------------------------------
__global__ __launch_bounds__(MODES) void ulayer_stitch(
    const float* __restrict__ diag_in,
    const float* __restrict__ theta,
    const float* __restrict__ phi,
    const int*   __restrict__ bs_m,
    const int*   __restrict__ bs_n,
    int nvars,
    float* __restrict__ U_re,
    float* __restrict__ U_im)
{
    extern __shared__ float smem[];
    float* Ure = smem;               // [128][128]
    float* Uim = smem + NTOT;        // [128][128]

    const int r = threadIdx.x;
    float* ur = Ure + r * MODES;
    float* ui = Uim + r * MODES;

    // U0 = diag(exp(i * diag))
    for (int c = 0; c < MODES; ++c) { ur[c] = 0.0f; ui[c] = 0.0f; }
    {
        float d = diag_in[r], sd, cd;
        sincosf(d, &sd, &cd);
        ur[r] = cd;
        ui[r] = sd;
    }

    // Sequential scan: new_m = e^{i phi} (cos*col_m + sin*col_n)
    //                  new_n = -sin*col_m + cos*col_n
    for (int s = 0; s < nvars; ++s) {
        const int m = bs_m[s];
        const int n = bs_n[s];
        float st, ct, sp, cp;
        sincosf(theta[s], &st, &ct);
        sincosf(phi[s],   &sp, &cp);

        const float umr = ur[m], umi = ui[m];
        const float unr = ur[n], uni = ui[n];

        const float tr = ct * umr + st * unr;   // (ct*um + st*un).re
        const float ti = ct * umi + st * uni;   // (ct*um + st*un).im
        ur[m] = cp * tr - sp * ti;              // * e^{i phi}
        ui[m] = cp * ti + sp * tr;
        ur[n] = ct * unr - st * umr;
        ui[n] = ct * uni - st * umi;
    }

    for (int c = 0; c < MODES; ++c) {
        U_re[r * MODES + c] = ur[c];
        U_im[r * MODES + c] = ui[c];
    }
}

// ---------------------------------------------------------------------------
// Kernel 2: y[b,i] = sum_j U[i,j] * x[b,j]   (complex, 4M method)
// C(B x 128) = X(B x 128) * B(128 x 128), B[k,n] = U[n,k]  (both row-major
// contiguous in k). f32 WMMA 16x16x4; each wave computes 16 batch rows x all
// 128 modes (two chunks of 64). U staged in LDS per block (128 KB).
// ---------------------------------------------------------------------------
__global__ __launch_bounds__(256) void ulayer_gemm(
    const float* __restrict__ x_re,
    const float* __restrict__ x_im,
    const float* __restrict__ U_re,
    const float* __restrict__ U_im,
    float* __restrict__ out)
{
    extern __shared__ float smem[];
    float* sUre = smem;          // [128][128], row n = U[n, 0..127]
    float* sUim = smem + NTOT;

    const int tid = threadIdx.x;

    // Stage U (re+im planes) into LDS with b128 copies.
    for (int idx = tid * 4; idx < NTOT; idx += 256 * 4) {
        *(float4*)&sUre[idx] = *(const float4*)&U_re[idx];
        *(float4*)&sUim[idx] = *(const float4*)&U_im[idx];
    }
    __syncthreads();

    const int lane = tid & 31;
    const int wave = tid >> 5;
    const int half = lane >> 4;     // 0: K+0..1, 1: K+2..3 (A/B fragment rule)
    const int l16  = lane & 15;

    const int mbase = blockIdx.x * 128 + wave * 16;   // this wave's batch rows
    const float* xr = x_re + (size_t)(mbase + l16) * MODES;
    const float* xi = x_im + (size_t)(mbase + l16) * MODES;

    for (int nb = 0; nb < MODES; nb += 64) {
        v8f accRe[4] = {};
        v8f accIm[4] = {};

        for (int k = 0; k < MODES; k += 4) {
            const int kk = k + 2 * half;
            const v2f aRe = *(const v2f*)(xr + kk);
            const v2f aIm = *(const v2f*)(xi + kk);

#pragma unroll
            for (int t = 0; t < 4; ++t) {
                const int n = nb + t * 16 + l16;
                const v2f bRe  = *(const v2f*)(sUre + n * MODES + kk);
                const v2f bIm  = *(const v2f*)(sUim + n * MODES + kk);
                const v2f bImN = -bIm;   // f32 WMMA has no A/B NEG modifier

                // acc_re += aRe*bRe - aIm*bIm ; acc_im += aRe*bIm + aIm*bRe
                accRe[t] = __builtin_amdgcn_wmma_f32_16x16x4_f32(
                    false, aRe, false, bRe, (short)0, accRe[t], false, false);
                accRe[t] = __builtin_amdgcn_wmma_f32_16x16x4_f32(
                    false, aIm, false, bImN, (short)0, accRe[t], false, false);
                accIm[t] = __builtin_amdgcn_wmma_f32_16x16x4_f32(
                    false, aRe, false, bIm, (short)0, accIm[t], false, false);
                accIm[t] = __builtin_amdgcn_wmma_f32_16x16x4_f32(
                    false, aIm, false, bRe, (short)0, accIm[t], false, false);
            }
        }

        // C/D layout: VGPR r -> M = r + 8*half, N = l16. Interleaved complex out.
#pragma unroll
        for (int t = 0; t < 4; ++t) {
            const int n = nb + t * 16 + l16;
#pragma unroll
            for (int r = 0; r < 8; ++r) {
                const int brow = mbase + r + 8 * half;
                v2f c;
                c.x = accRe[t][r];
                c.y = accIm[t][r];
                *(v2f*)(out + ((size_t)brow * MODES + n) * 2) = c;
            }
        }
    }
}

// ---------------------------------------------------------------------------
extern "C" void kernel_launch(void* const* d_in, const int* in_sizes, int n_in,
                              void* d_out, int out_size, void* d_ws, size_t ws_size,
                              hipStream_t stream) {
    const float* x_re  = (const float*)d_in[0];
    const float* x_im  = (const float*)d_in[1];
    const float* diag  = (const float*)d_in[2];
    const float* theta = (const float*)d_in[3];
    const float* phi   = (const float*)d_in[4];
    const int*   bs_m  = (const int*)d_in[5];
    const int*   bs_n  = (const int*)d_in[6];
    const int nvars = in_sizes[3];
    const int batch = in_sizes[0] / MODES;

    float* U_re = (float*)d_ws;
    float* U_im = U_re + NTOT;

    const size_t lds_bytes = 2u * NTOT * sizeof(float);   // 128 KB (<320 KB WGP LDS)
    hipFuncSetAttribute((const void*)ulayer_stitch,
                        hipFuncAttributeMaxDynamicSharedMemorySize, (int)lds_bytes);
    hipFuncSetAttribute((const void*)ulayer_gemm,
                        hipFuncAttributeMaxDynamicSharedMemorySize, (int)lds_bytes);

    ulayer_stitch<<<1, MODES, lds_bytes, stream>>>(diag, theta, phi, bs_m, bs_n,
                                                   nvars, U_re, U_im);

    const int grid = batch / 128;   // 512 blocks of 8 waves x 16 rows
    ulayer_gemm<<<grid, 256, lds_bytes, stream>>>(x_re, x_im, U_re, U_im,
                                                  (float*)d_out);
}